// CtcLoss_83313775608583
// MI455X (gfx1250) — compile-verified
//
#include <hip/hip_runtime.h>

// ---------------------------------------------------------------------------
// CTC batch cost, MI455X (gfx1250, wave32).
//   B=128, T=1024, C=256 (blank=C-1), L=128, S=2L+1=257.
// One wave per batch sample. Lane l owns 9 contiguous lattice states
// s = 9l..9l+8 (288 >= 257). Forward DP runs in log2 domain so the
// transcendentals are single native v_exp_f32 / v_log_f32 ops.
// Probability rows are prefetched into an LDS ring with CDNA5
// global_load_async_to_lds_b128 (ASYNCcnt), depth 8.
// ---------------------------------------------------------------------------

#define GLOBAL_AS __attribute__((address_space(1)))
#define LDS_AS    __attribute__((address_space(3)))

typedef __attribute__((__vector_size__(4 * sizeof(int)))) int v4i;
typedef GLOBAL_AS v4i* gv4i_p;   // global (addrspace 1) int4*
typedef LDS_AS    v4i* lv4i_p;   // LDS    (addrspace 3) int4*

#ifndef __has_builtin
#define __has_builtin(x) 0
#endif

#if __has_builtin(__builtin_amdgcn_global_load_async_to_lds_b128)
#define USE_ASYNC 1
#else
#define USE_ASYNC 0
#endif

#if __has_builtin(__builtin_amdgcn_s_wait_asynccnt)
#define WAIT_ASYNC(n) __builtin_amdgcn_s_wait_asynccnt(n)
#else
#define WAIT_ASYNC(n) asm volatile("s_wait_asynccnt " #n ::: "memory")
#endif

#if __has_builtin(__builtin_amdgcn_exp2f)
#define EXP2F(x) __builtin_amdgcn_exp2f(x)
#else
#define EXP2F(x) exp2f(x)
#endif
#if __has_builtin(__builtin_amdgcn_logf)
#define LOG2F(x) __builtin_amdgcn_logf(x)   // v_log_f32 == log2
#else
#define LOG2F(x) __log2f(x)
#endif

#define NEGF  (-1.0e9f)
#define EPSF  (1.0e-7f)
#define LN2F  (0.6931471805599453f)

#define B_    128
#define T_    1024
#define C_    256
#define L_    128
#define SLOTS 9        // states per lane (contiguous)
#define DEPTH 8        // LDS prefetch ring depth (rows of C_ floats)

// async-copy one 1KB probability row (256 floats) into an LDS ring slot:
// 32 lanes x 16B x 2 instructions, ASYNCcnt += 2
__device__ __forceinline__ void async_row(const float* gp, float* lp0) {
    __builtin_amdgcn_global_load_async_to_lds_b128(
        (gv4i_p)(GLOBAL_AS float*)gp, (lv4i_p)(LDS_AS float*)lp0, 0, 0);
    __builtin_amdgcn_global_load_async_to_lds_b128(
        (gv4i_p)(GLOBAL_AS float*)(gp + 128), (lv4i_p)(LDS_AS float*)(lp0 + 128), 0, 0);
}

// log2-domain logaddexp of three terms
__device__ __forceinline__ float lae3_2(float x, float y, float z) {
    float m = fmaxf(x, fmaxf(y, z));
    float s = EXP2F(x - m) + EXP2F(y - m) + EXP2F(z - m);
    return m + LOG2F(s);
}

extern "C" __global__ __launch_bounds__(32)
void ctc_loss_kernel(const float* __restrict__ y,       // [B,T,C] probs
                     const int*   __restrict__ labels,  // [B,L]
                     const int*   __restrict__ ilen,    // [B]
                     const int*   __restrict__ llen,    // [B]
                     float*       __restrict__ out)     // [B]
{
    __shared__ __align__(16) float ring[DEPTH * C_];   // 8 KB prob-row ring
    __shared__ float shA[SLOTS * 32];                  // final alpha dump (288)

    const int b    = blockIdx.x;
    const int lane = threadIdx.x;                      // 0..31 (wave32)
    int Tin = ilen[b];  if (Tin > T_) Tin = T_;
    const int len    = llen[b];
    const int Svalid = 2 * len + 1;

    const float* __restrict__ yb = y + (size_t)b * T_ * C_;
    const int*   __restrict__ lb = labels + b * L_;

    // --- per-lane lattice metadata (ext label column, validity, skip-ok) ---
    int  ecol[SLOTS];
    bool valid[SLOTS];
    bool skipok[SLOTS];
#pragma unroll
    for (int j = 0; j < SLOTS; ++j) {
        const int s  = SLOTS * lane + j;
        int  col = C_ - 1;                 // blank
        bool sk  = false;
        if ((s & 1) && s < 2 * L_) {       // odd s -> label position (s-1)/2
            const int li = (s - 1) >> 1;
            col = lb[li];
            const int prev = (li >= 1) ? lb[li - 1] : -1;
            sk = (s >= 2) && (col != prev);
        }
        ecol[j]   = col;
        valid[j]  = (s < Svalid);
        skipok[j] = sk;
    }

    float a[SLOTS];
#pragma unroll
    for (int j = 0; j < SLOTS; ++j) a[j] = NEGF;

#if USE_ASYNC
    // prologue: launch async copies of rows 0..DEPTH-1 into the LDS ring
    for (int i = 0; i < DEPTH; ++i)
        async_row(yb + (size_t)i * C_ + lane * 4, &ring[i * C_ + lane * 4]);
#endif

    const int rotAddr = ((lane + 31) & 31) << 2;   // ds_bpermute: lane-1 (wraps)

    for (int t = 0; t < Tin; ++t) {
        // ---- obtain this timestep's gathered probabilities ----
        float g[SLOTS];
#if USE_ASYNC
        WAIT_ASYNC(14);                            // 2*(DEPTH-1): row t complete
        asm volatile("" ::: "memory");
        const float* row = &ring[(t & (DEPTH - 1)) * C_];
#else
        const float* row = yb + (size_t)t * C_;
#endif
#pragma unroll
        for (int j = 0; j < SLOTS; ++j) g[j] = row[ecol[j]];   // ds/global gather

        float lpv[SLOTS];
#pragma unroll
        for (int j = 0; j < SLOTS; ++j) lpv[j] = LOG2F(g[j] + EPSF);

        if (t == 0) {
            // alpha0: only s=0 (blank) and s=1 (first label) live, both on lane 0
#pragma unroll
            for (int j = 0; j < SLOTS; ++j)
                a[j] = (lane == 0 && j < 2) ? lpv[j] : NEGF;
        } else {
            // cross-lane neighbors: states 9l-1 and 9l-2 live in lane l-1 slots 8,7
            float n8 = __int_as_float(__builtin_amdgcn_ds_bpermute(rotAddr, __float_as_int(a[8])));
            float n7 = __int_as_float(__builtin_amdgcn_ds_bpermute(rotAddr, __float_as_int(a[7])));
            const float p1_0 = (lane == 0) ? NEGF : n8;   // s-1 for j=0
            const float p2_0 = (lane == 0) ? NEGF : n7;   // s-2 for j=0
            const float p2_1 = (lane == 0) ? NEGF : n8;   // s-2 for j=1

            float na[SLOTS];
#pragma unroll
            for (int j = 0; j < SLOTS; ++j) {
                const float p1 = (j == 0) ? p1_0 : a[j - 1];
                const float p2 = (j == 0) ? p2_0 : ((j == 1) ? p2_1 : a[j - 2]);
                const float z  = skipok[j] ? p2 : NEGF;
                na[j] = valid[j] ? (lae3_2(a[j], p1, z) + lpv[j]) : NEGF;
            }
#pragma unroll
            for (int j = 0; j < SLOTS; ++j) a[j] = na[j];
        }

#if USE_ASYNC
        // refill the just-consumed ring slot with row t+DEPTH
        const int tn = t + DEPTH;
        if (tn < Tin) {
            asm volatile("" ::: "memory");         // keep gathers before refill
            async_row(yb + (size_t)tn * C_ + lane * 4,
                      &ring[(tn & (DEPTH - 1)) * C_ + lane * 4]);
        }
#endif
    }
    // (alpha is frozen for t >= input_length in the reference -> early exit OK)

    // ---- readout: loss = -ln2 * log2addexp(alpha[2len], alpha[2len-1]) ----
#pragma unroll
    for (int j = 0; j < SLOTS; ++j) shA[SLOTS * lane + j] = a[j];
    __syncthreads();
    if (lane == 0) {
        const int e1 = 2 * len;
        const int e0 = (e1 > 0) ? (e1 - 1) : 0;
        const float l1 = shA[e1];
        const float l2 = shA[e0];
        const float m  = fmaxf(l1, l2);
        const float lae = m + LOG2F(EXP2F(l1 - m) + EXP2F(l2 - m));
        out[b] = -LN2F * lae;
    }
}

extern "C" void kernel_launch(void* const* d_in, const int* in_sizes, int n_in,
                              void* d_out, int out_size, void* d_ws, size_t ws_size,
                              hipStream_t stream) {
    (void)n_in; (void)out_size; (void)d_ws; (void)ws_size;
    const float* y      = (const float*)d_in[0];   // [B,T,C] float32
    const int*   labels = (const int*)d_in[1];     // [B,L]   int32
    const int*   ilen   = (const int*)d_in[2];     // [B,1]   int32
    const int*   llen   = (const int*)d_in[3];     // [B,1]   int32
    float*       out    = (float*)d_out;           // [B,1]   float32

    const int B = in_sizes[2];                     // 128
    ctc_loss_kernel<<<dim3(B), dim3(32), 0, stream>>>(y, labels, ilen, llen, out);
}